// ActionModel_59141699666058
// MI455X (gfx1250) — compile-verified
//
#include <hip/hip_runtime.h>

typedef float v2f __attribute__((ext_vector_type(2)));
typedef float v8f __attribute__((ext_vector_type(8)));

#define HD   128   // hidden / feature width (F == H == 128)
#define EDIM 32    // edge feature dim
#define NGRAPH 64
#define NACT 16

// ---------------- utility ----------------
__global__ void zero_kernel(float* __restrict__ p, int n) {
    int i = blockIdx.x * blockDim.x + threadIdx.x;
    if (i < n) p[i] = 0.0f;
}

// ---------------- fused edge message kernel ----------------
// For one 16-edge tile (blockIdx.x) each of 8 waves owns a 16-channel tile:
//   msg = relu(x[src] + edge_attr @ We^T + be)   then atomicAdd into agg[dst].
// edge-lin done with 8x V_WMMA_F32_16X16X4_F32 (K=32).
__global__ void edge_msg_kernel(const float* __restrict__ xin,
                                const float* __restrict__ ea,    // [E,32]
                                const float* __restrict__ We,    // [128,32]
                                const float* __restrict__ be,    // [128]
                                const int*   __restrict__ src,
                                const int*   __restrict__ dst,
                                float*       __restrict__ agg)   // [N,128]
{
    const int etile = blockIdx.x;
    const int wv    = threadIdx.x >> 5;   // 0..7 -> channel tile
    const int lane  = threadIdx.x & 31;
    const int hi    = lane >> 4;          // half-wave select
    const int lo    = lane & 15;

    const int arow = etile * 16 + lo;     // edge row for A fragment
    const int bcol = wv * 16 + lo;        // output channel for B fragment

    v8f acc = {0.f, 0.f, 0.f, 0.f, 0.f, 0.f, 0.f, 0.f};
    #pragma unroll
    for (int kk = 0; kk < EDIM / 4; ++kk) {
        const int k0 = kk * 4 + 2 * hi;
        v2f a, b;
        a.x = ea[arow * EDIM + k0];
        a.y = ea[arow * EDIM + k0 + 1];
        b.x = We[bcol * EDIM + k0];       // B[k][n] = We[n][k]
        b.y = We[bcol * EDIM + k0 + 1];
        acc = __builtin_amdgcn_wmma_f32_16x16x4_f32(false, a, false, b,
                                                    (short)0, acc, false, false);
    }

    const int   c    = wv * 16 + lo;      // C-fragment column == channel
    const float bias = be[c];
    #pragma unroll
    for (int r = 0; r < 8; ++r) {
        const int m = etile * 16 + r + 8 * hi;   // edge index of this element
        const int s = src[m];
        const int d = dst[m];
        float v = acc[r] + bias + xin[s * HD + c];
        v = fmaxf(v, 0.0f);
        atomicAdd(&agg[d * HD + c], v);
    }
}

// ---------------- WMMA GEMM: C[M,128] = (A1 (+A2)) @ W^T + bias ----------------
// grid.x = M/16 row tiles, 8 waves per block cover the 8 output-channel tiles.
// K and the residual-add flag are compile-time so the 32-step K-loop fully
// unrolls into a straight line of v_wmma_f32_16x16x4_f32 with batched loads.
template <int K, bool ADD>
__global__ void gemm_wmma_kernel(const float* __restrict__ A1,
                                 const float* __restrict__ A2,   // used iff ADD
                                 const float* __restrict__ W,    // [128,K] row-major
                                 const float* __restrict__ bias, // [128]
                                 float*       __restrict__ C)    // [M,128]
{
    const int mtile = blockIdx.x;
    const int wv    = threadIdx.x >> 5;
    const int lane  = threadIdx.x & 31;
    const int hi    = lane >> 4;
    const int lo    = lane & 15;

    const int arow = mtile * 16 + lo;
    const int bcol = wv * 16 + lo;

    v8f acc = {0.f, 0.f, 0.f, 0.f, 0.f, 0.f, 0.f, 0.f};
    #pragma unroll
    for (int kk = 0; kk < K / 4; ++kk) {
        const int k0 = kk * 4 + 2 * hi;
        v2f a, b;
        a.x = A1[arow * K + k0];
        a.y = A1[arow * K + k0 + 1];
        if (ADD) {
            a.x += A2[arow * K + k0];
            a.y += A2[arow * K + k0 + 1];
        }
        b.x = W[bcol * K + k0];
        b.y = W[bcol * K + k0 + 1];
        acc = __builtin_amdgcn_wmma_f32_16x16x4_f32(false, a, false, b,
                                                    (short)0, acc, false, false);
    }

    const int   c  = wv * 16 + lo;
    const float bb = bias[c];
    #pragma unroll
    for (int r = 0; r < 8; ++r) {
        const int row = mtile * 16 + r + 8 * hi;
        C[row * HD + c] = acc[r] + bb;
    }
}

// ---------------- column-wise sum / sum-of-squares for train-mode BN ----------------
__global__ void col_stats_kernel(const float* __restrict__ X, int M,
                                 float* __restrict__ sums, float* __restrict__ sumsq)
{
    const int c  = threadIdx.x;           // 0..127 (coalesced across channels)
    const int r0 = blockIdx.x * 64;
    const int r1 = (r0 + 64 < M) ? (r0 + 64) : M;
    float s = 0.f, s2 = 0.f;
    for (int r = r0; r < r1; ++r) {
        const float v = X[r * HD + c];
        s  += v;
        s2 += v * v;
    }
    atomicAdd(&sums[c], s);
    atomicAdd(&sumsq[c], s2);
}

// ---------------- BN (biased batch stats) + ReLU (+ optional sigmoid) ----------------
__global__ void bn_apply_kernel(const float* __restrict__ X,
                                const float* __restrict__ sums,
                                const float* __restrict__ sumsq,
                                const float* __restrict__ gamma,
                                const float* __restrict__ beta,
                                int M, int do_sigmoid,
                                float* __restrict__ Y)
{
    const int i = blockIdx.x * blockDim.x + threadIdx.x;
    if (i >= M * HD) return;
    const int   c    = i & (HD - 1);
    const float inv  = 1.0f / (float)M;
    const float mean = sums[c] * inv;
    const float var  = sumsq[c] * inv - mean * mean;
    float y = (X[i] - mean) * rsqrtf(var + 1e-5f) * gamma[c] + beta[c];
    y = fmaxf(y, 0.0f);
    if (do_sigmoid) y = 1.0f / (1.0f + expf(-y));
    Y[i] = y;
}

// ---------------- per-graph mean pooling ----------------
__global__ void pool_sum_kernel(const float* __restrict__ h,
                                const int*   __restrict__ batch,
                                float* __restrict__ pooled,   // [64,128]
                                float* __restrict__ cnt,      // [64]
                                int M)
{
    const int i = blockIdx.x * blockDim.x + threadIdx.x;
    if (i >= M * HD) return;
    const int n = i >> 7;
    const int c = i & (HD - 1);
    const int g = batch[n];
    atomicAdd(&pooled[g * HD + c], h[i]);
    if (c == 0) atomicAdd(&cnt[g], 1.0f);
}

__global__ void pool_div_kernel(float* __restrict__ pooled,
                                const float* __restrict__ cnt)
{
    const int i = blockIdx.x * blockDim.x + threadIdx.x;
    if (i >= NGRAPH * HD) return;
    pooled[i] /= fmaxf(cnt[i >> 7], 1.0f);
}

// ---------------- final action head: sigmoid(z @ A3w^T + A3b) ----------------
__global__ void final_kernel(const float* __restrict__ z,   // [64,128]
                             const float* __restrict__ W,   // [16,128]
                             const float* __restrict__ b,   // [16]
                             float* __restrict__ out)       // [64,16]
{
    const int i = blockIdx.x * blockDim.x + threadIdx.x;
    if (i >= NGRAPH * NACT) return;
    const int g = i >> 4;
    const int a = i & (NACT - 1);
    float s = b[a];
    #pragma unroll 4
    for (int k = 0; k < HD; ++k) s += z[g * HD + k] * W[a * HD + k];
    out[i] = 1.0f / (1.0f + expf(-s));
}

// ---------------- host launch ----------------
extern "C" void kernel_launch(void* const* d_in, const int* in_sizes, int n_in,
                              void* d_out, int out_size, void* d_ws, size_t ws_size,
                              hipStream_t stream)
{
    // setup_inputs() dict order
    const float* x     = (const float*)d_in[0];
    const float* ea    = (const float*)d_in[1];
    const int*   ei    = (const int*)  d_in[2];
    const int*   batch = (const int*)  d_in[3];
    const float* W1  = (const float*)d_in[4];  const float* b1  = (const float*)d_in[5];
    const float* We1 = (const float*)d_in[6];  const float* be1 = (const float*)d_in[7];
    const float* g1  = (const float*)d_in[8];  const float* bt1 = (const float*)d_in[9];
    const float* W2  = (const float*)d_in[10]; const float* b2  = (const float*)d_in[11];
    const float* We2 = (const float*)d_in[12]; const float* be2 = (const float*)d_in[13];
    const float* g2  = (const float*)d_in[14]; const float* bt2 = (const float*)d_in[15];
    const float* A1w = (const float*)d_in[16]; const float* A1b = (const float*)d_in[17];
    const float* Ag1 = (const float*)d_in[18]; const float* Ab1 = (const float*)d_in[19];
    const float* A2w = (const float*)d_in[20]; const float* A2b = (const float*)d_in[21];
    const float* Ag2 = (const float*)d_in[22]; const float* Ab2 = (const float*)d_in[23];
    const float* A3w = (const float*)d_in[24]; const float* A3b = (const float*)d_in[25];

    const int N = in_sizes[0] / HD;     // 50000
    const int E = in_sizes[2] / 2;      // 800000
    const int* src = ei;
    const int* dst = ei + E;

    const int NF = N * HD;
    float* ws     = (float*)d_ws;
    float* agg    = ws;                 // [N,128]
    float* lin    = agg + NF;           // [N,128] (also h2 in-place)
    float* h1     = lin + NF;           // [N,128]
    float* sums   = h1 + NF;            // [128]
    float* sumsq  = sums + HD;          // [128]
    float* pooled = sumsq + HD;         // [64,128]
    float* cnt    = pooled + NGRAPH * HD;  // [64]
    float* z1     = cnt + NGRAPH;       // [64,128]
    float* z2     = z1 + NGRAPH * HD;   // [64,128]

    #define ZERO(p, n) zero_kernel<<<((n) + 255) / 256, 256, 0, stream>>>((p), (n))

    // ---- GINE layer 1 ----
    ZERO(agg, NF);
    edge_msg_kernel<<<E / 16, 256, 0, stream>>>(x, ea, We1, be1, src, dst, agg);
    gemm_wmma_kernel<HD, true><<<N / 16, 256, 0, stream>>>(x, agg, W1, b1, lin);
    ZERO(sums, 2 * HD);
    col_stats_kernel<<<(N + 63) / 64, HD, 0, stream>>>(lin, N, sums, sumsq);
    bn_apply_kernel<<<(NF + 255) / 256, 256, 0, stream>>>(lin, sums, sumsq, g1, bt1, N, 0, h1);

    // ---- GINE layer 2 ----
    ZERO(agg, NF);
    edge_msg_kernel<<<E / 16, 256, 0, stream>>>(h1, ea, We2, be2, src, dst, agg);
    gemm_wmma_kernel<HD, true><<<N / 16, 256, 0, stream>>>(h1, agg, W2, b2, lin);
    ZERO(sums, 2 * HD);
    col_stats_kernel<<<(N + 63) / 64, HD, 0, stream>>>(lin, N, sums, sumsq);
    bn_apply_kernel<<<(NF + 255) / 256, 256, 0, stream>>>(lin, sums, sumsq, g2, bt2, N, 1, lin);

    // ---- per-graph mean pooling ----
    ZERO(pooled, NGRAPH * HD + NGRAPH);   // pooled + cnt contiguous
    pool_sum_kernel<<<(NF + 255) / 256, 256, 0, stream>>>(lin, batch, pooled, cnt, N);
    pool_div_kernel<<<(NGRAPH * HD + 255) / 256, 256, 0, stream>>>(pooled, cnt);

    // ---- action MLP layer 1 ----
    gemm_wmma_kernel<HD, false><<<NGRAPH / 16, 256, 0, stream>>>(pooled, nullptr, A1w, A1b, z1);
    ZERO(sums, 2 * HD);
    col_stats_kernel<<<1, HD, 0, stream>>>(z1, NGRAPH, sums, sumsq);
    bn_apply_kernel<<<(NGRAPH * HD + 255) / 256, 256, 0, stream>>>(z1, sums, sumsq, Ag1, Ab1, NGRAPH, 0, z1);

    // ---- action MLP layer 2 ----
    gemm_wmma_kernel<HD, false><<<NGRAPH / 16, 256, 0, stream>>>(z1, nullptr, A2w, A2b, z2);
    ZERO(sums, 2 * HD);
    col_stats_kernel<<<1, HD, 0, stream>>>(z2, NGRAPH, sums, sumsq);
    bn_apply_kernel<<<(NGRAPH * HD + 255) / 256, 256, 0, stream>>>(z2, sums, sumsq, Ag2, Ab2, NGRAPH, 0, z2);

    // ---- final sigmoid head ----
    final_kernel<<<(NGRAPH * NACT + 255) / 256, 256, 0, stream>>>(z2, A3w, A3b, (float*)d_out);

    #undef ZERO
}